// FactorGraphLayer_75788992905474
// MI455X (gfx1250) — compile-verified
//
#include <hip/hip_runtime.h>
#include <math.h>

// ---------------------------------------------------------------------------
// FactorGraphLayer refinement, MI455X (gfx1250).
// Memory-bound op: ~1 flop/byte -> HBM roofline (23.3 TB/s) dominates.
// Strategy: closed-form reformulation (uniform rescale + renorm preserves
// ratios => carry only scale products), keep 400KB node arrays + atomics in
// L2, prefetch streaming edge data, NT stores for the 64MB streaming output.
// ---------------------------------------------------------------------------

#define N_NODES 100000
#define N_EDGES 3200000
#define NODE_C  7
#define EDGE_C  5
#define GAMMA   1.0f
#define EPS_DEG 1e-6f
#define EPS_LOG 1e-9f

#define EDGE_BLOCKS 2048
#define TPB 256   // 8 wave32 waves per block

// ---- ws layout (floats) ----------------------------------------------------
// [0]=avgE, [1]=avgN, pad to 16
// na0[N_NODES], na1[N_NODES], sum1[N_NODES], sum2[N_NODES], deg[N_NODES],
// nprob[N_NODES*NODE_C]
// total = 16 + 5*100000 + 700000 floats = 4.8 MB

__global__ void k_avg_factors(const float* __restrict__ nfw,
                              const float* __restrict__ efw,
                              float* __restrict__ avg) {
    if (blockIdx.x == 0 && threadIdx.x == 0) {
        float se = 0.f;
        for (int r = 1; r < EDGE_C; ++r)
            for (int c = 1; c < NODE_C; ++c) se += efw[r * NODE_C + c];
        avg[0] = se / (float)((EDGE_C - 1) * (NODE_C - 1));
        float sn = 0.f;
        for (int r = 1; r < NODE_C; ++r)
            for (int c = 1; c < EDGE_C; ++c) sn += nfw[r * EDGE_C + c];
        avg[1] = sn / (float)((NODE_C - 1) * (EDGE_C - 1));
    }
}

__global__ void k_node_init(const float* __restrict__ logits,
                            float* __restrict__ nprob,
                            float* __restrict__ na0,
                            float* __restrict__ sum1,
                            float* __restrict__ sum2,
                            float* __restrict__ deg) {
    int n = blockIdx.x * blockDim.x + threadIdx.x;
    if (n >= N_NODES) return;
    float l[NODE_C];
    float m = -INFINITY;
    #pragma unroll
    for (int i = 0; i < NODE_C; ++i) { l[i] = logits[n * NODE_C + i]; m = fmaxf(m, l[i]); }
    float s = 0.f;
    #pragma unroll
    for (int i = 0; i < NODE_C; ++i) { l[i] = expf(l[i] - m); s += l[i]; }
    float inv = 1.f / s;
    #pragma unroll
    for (int i = 0; i < NODE_C; ++i) nprob[n * NODE_C + i] = l[i] * inv;
    na0[n]  = 1.f - l[0] * inv;
    sum1[n] = 0.f;   // deterministic re-init every launch
    sum2[n] = 0.f;
    deg[n]  = 0.f;
}

// Edge pass 1: softmax + first rescale in closed form; scatter abnormality and
// degree into L2-resident node accumulators. Nothing written per-edge to HBM.
__global__ void k_edge_pass1(const float* __restrict__ elog,
                             const int*   __restrict__ eidx,
                             const float* __restrict__ na0,
                             const float* __restrict__ avg,
                             float* __restrict__ sum1,
                             float* __restrict__ deg) {
    const float avgE   = avg[0];
    const int   stride = gridDim.x * blockDim.x;
    for (int e = blockIdx.x * blockDim.x + threadIdx.x; e < N_EDGES; e += stride) {
        int en = e + stride;                       // stream prefetch one stride ahead
        if (en < N_EDGES) {
            __builtin_prefetch(&elog[(size_t)en * EDGE_C], 0, 3);  // global_prefetch_b8
            __builtin_prefetch(&eidx[en], 0, 3);
            __builtin_prefetch(&eidx[N_EDGES + en], 0, 3);
        }
        const int s = eidx[e];
        const int d = eidx[N_EDGES + e];
        const float a = fmaxf(na0[s], na0[d]);     // gathers hit L2 (400KB array)
        const float* row = elog + (size_t)e * EDGE_C;
        float l0 = row[0], l1 = row[1], l2 = row[2], l3 = row[3], l4 = row[4];
        float m  = fmaxf(fmaxf(fmaxf(l0, l1), fmaxf(l2, l3)), l4);
        float e0 = expf(l0 - m);
        float S  = expf(l1 - m) + expf(l2 - m) + expf(l3 - m) + expf(l4 - m);
        float F  = 1.f + GAMMA * avgE * a;
        float abn = F * S / (e0 + F * S);          // 1 - p0 after rescale+renorm
        atomicAdd(&sum1[s], abn);
        atomicAdd(&sum1[d], abn);
        atomicAdd(&deg[s], 1.f);
        atomicAdd(&deg[d], 1.f);
    }
}

__global__ void k_node_update(float* __restrict__ nprob,
                              const float* __restrict__ sum,
                              const float* __restrict__ deg,
                              const float* __restrict__ avg,
                              float* __restrict__ na_out) {
    int n = blockIdx.x * blockDim.x + threadIdx.x;
    if (n >= N_NODES) return;
    float mean = sum[n] / (deg[n] + EPS_DEG);
    float f = 1.f + GAMMA * avg[1] * mean;
    float p[NODE_C];
    #pragma unroll
    for (int i = 0; i < NODE_C; ++i) p[i] = nprob[n * NODE_C + i];
    float tail = 0.f;
    #pragma unroll
    for (int i = 1; i < NODE_C; ++i) tail += p[i];
    float invZ = 1.f / (p[0] + f * tail);
    nprob[n * NODE_C] = p[0] * invZ;
    #pragma unroll
    for (int i = 1; i < NODE_C; ++i) nprob[n * NODE_C + i] = p[i] * f * invZ;
    na_out[n] = 1.f - p[0] * invZ;
}

// Edge pass 2: recompute F1 from cached na0 (no 12.8MB spill), apply F1*F2 in
// closed form, emit final log-probs with non-temporal stores (keep L2 for the
// logits/atomics), scatter abnormality for the final node update.
__global__ void k_edge_pass2(const float* __restrict__ elog,
                             const int*   __restrict__ eidx,
                             const float* __restrict__ na0,
                             const float* __restrict__ na1,
                             const float* __restrict__ avg,
                             float* __restrict__ sum2,
                             float* __restrict__ out_edge) {
    const float avgE   = avg[0];
    const int   stride = gridDim.x * blockDim.x;
    for (int e = blockIdx.x * blockDim.x + threadIdx.x; e < N_EDGES; e += stride) {
        int en = e + stride;
        if (en < N_EDGES) {
            __builtin_prefetch(&elog[(size_t)en * EDGE_C], 0, 3);
            __builtin_prefetch(&eidx[en], 0, 3);
            __builtin_prefetch(&eidx[N_EDGES + en], 0, 3);
        }
        const int s = eidx[e];
        const int d = eidx[N_EDGES + e];
        float F1 = 1.f + GAMMA * avgE * fmaxf(na0[s], na0[d]);
        float F2 = 1.f + GAMMA * avgE * fmaxf(na1[s], na1[d]);
        float F  = F1 * F2;
        const float* row = elog + (size_t)e * EDGE_C;
        float l0 = row[0], l1 = row[1], l2 = row[2], l3 = row[3], l4 = row[4];
        float m  = fmaxf(fmaxf(fmaxf(l0, l1), fmaxf(l2, l3)), l4);
        float e0 = expf(l0 - m);
        float e1 = expf(l1 - m), e2 = expf(l2 - m), e3 = expf(l3 - m), e4 = expf(l4 - m);
        float invZ = 1.f / (e0 + F * (e1 + e2 + e3 + e4));
        float p0 = e0 * invZ;
        float* o = out_edge + (size_t)e * EDGE_C;
        __builtin_nontemporal_store(logf(p0 + EPS_LOG),          &o[0]);
        __builtin_nontemporal_store(logf(F * e1 * invZ + EPS_LOG), &o[1]);
        __builtin_nontemporal_store(logf(F * e2 * invZ + EPS_LOG), &o[2]);
        __builtin_nontemporal_store(logf(F * e3 * invZ + EPS_LOG), &o[3]);
        __builtin_nontemporal_store(logf(F * e4 * invZ + EPS_LOG), &o[4]);
        float abn = 1.f - p0;
        atomicAdd(&sum2[s], abn);
        atomicAdd(&sum2[d], abn);
    }
}

__global__ void k_node_final(const float* __restrict__ nprob,
                             const float* __restrict__ sum,
                             const float* __restrict__ deg,
                             const float* __restrict__ avg,
                             float* __restrict__ out_node) {
    int n = blockIdx.x * blockDim.x + threadIdx.x;
    if (n >= N_NODES) return;
    float mean = sum[n] / (deg[n] + EPS_DEG);
    float f = 1.f + GAMMA * avg[1] * mean;
    float p[NODE_C];
    #pragma unroll
    for (int i = 0; i < NODE_C; ++i) p[i] = nprob[n * NODE_C + i];
    float tail = 0.f;
    #pragma unroll
    for (int i = 1; i < NODE_C; ++i) tail += p[i];
    float invZ = 1.f / (p[0] + f * tail);
    out_node[n * NODE_C] = logf(p[0] * invZ + EPS_LOG);
    #pragma unroll
    for (int i = 1; i < NODE_C; ++i)
        out_node[n * NODE_C + i] = logf(p[i] * f * invZ + EPS_LOG);
}

extern "C" void kernel_launch(void* const* d_in, const int* in_sizes, int n_in,
                              void* d_out, int out_size, void* d_ws, size_t ws_size,
                              hipStream_t stream) {
    const float* node_logits = (const float*)d_in[0];   // [100000,7]
    const float* edge_logits = (const float*)d_in[1];   // [3200000,5]
    const int*   edge_index  = (const int*)  d_in[2];   // [2,3200000]
    const float* nfw         = (const float*)d_in[3];   // [7,5]
    const float* efw         = (const float*)d_in[4];   // [5,7]

    float* ws    = (float*)d_ws;
    float* avg   = ws;                      // [0]=avgE, [1]=avgN
    float* na0   = ws + 16;
    float* na1   = na0  + N_NODES;
    float* sum1  = na1  + N_NODES;
    float* sum2  = sum1 + N_NODES;
    float* deg   = sum2 + N_NODES;
    float* nprob = deg  + N_NODES;          // N_NODES*NODE_C

    float* out_node = (float*)d_out;                     // 700000 floats
    float* out_edge = out_node + (size_t)N_NODES * NODE_C; // 16000000 floats

    const int nodeBlocks = (N_NODES + TPB - 1) / TPB;

    k_avg_factors<<<1, 64, 0, stream>>>(nfw, efw, avg);
    k_node_init  <<<nodeBlocks, TPB, 0, stream>>>(node_logits, nprob, na0, sum1, sum2, deg);
    // iteration 1
    k_edge_pass1 <<<EDGE_BLOCKS, TPB, 0, stream>>>(edge_logits, edge_index, na0, avg, sum1, deg);
    k_node_update<<<nodeBlocks, TPB, 0, stream>>>(nprob, sum1, deg, avg, na1);
    // iteration 2 (final edge probs -> logs fused here)
    k_edge_pass2 <<<EDGE_BLOCKS, TPB, 0, stream>>>(edge_logits, edge_index, na0, na1, avg, sum2, out_edge);
    k_node_final <<<nodeBlocks, TPB, 0, stream>>>(nprob, sum2, deg, avg, out_node);
}